// IVPLoss_23922967839485
// MI455X (gfx1250) — compile-verified
//
#include <hip/hip_runtime.h>

#define HH 512
#define WW 512
#define HW (HH * WW)
#define NB 8
#define NSTEPS 8
#define DXC 0.5f
// total trajectory elements = (NSTEPS+1) * B * 2 * H * W = 9*8*2*512*512
#define INV_COUNT (1.0f / 37748736.0f)
#define NPARTIALS 8192

typedef __attribute__((ext_vector_type(2))) float v2f;
typedef __attribute__((ext_vector_type(4))) float v4f;
typedef __attribute__((ext_vector_type(8))) float v8f;

// Bilinear sample of both channels of one (2,512,512) field at (px,py),
// clipping to the image domain (matches reference _bilinear_sample).
// Pair trick: xp = min(floor(cx),510), wx = cx-xp reproduces the reference's
// x1 = min(x0+1,511) clip exactly (at cx==511, wx==1 selects v[511]),
// letting each (v0,v1) pair load as one dword-aligned b64.
__device__ __forceinline__ void bilerp2(const float* __restrict__ base,
                                        float px, float py,
                                        float& vx, float& vy) {
    float cx = fminf(fmaxf(px, 0.0f), 511.0f);
    float cy = fminf(fmaxf(py, 0.0f), 511.0f);
    int xp = min((int)floorf(cx), 510);
    int yp = min((int)floorf(cy), 510);
    float wx = cx - (float)xp;
    float wy = cy - (float)yp;
    int r0 = (yp << 9) + xp;       // row yp, col xp
    int r1 = r0 + WW;              // row yp+1
    v2f a0 = *(const v2f*)(base + r0);        // ch0: (v00, v01)
    v2f a1 = *(const v2f*)(base + r1);        // ch0: (v10, v11)
    v2f b0 = *(const v2f*)(base + HW + r0);   // ch1: (v00, v01)
    v2f b1 = *(const v2f*)(base + HW + r1);   // ch1: (v10, v11)
    float t0 = fmaf(wx, a0.y - a0.x, a0.x);
    float t1 = fmaf(wx, a1.y - a1.x, a1.x);
    vx = fmaf(wy, t1 - t0, t0);
    float s0 = fmaf(wx, b0.y - b0.x, b0.x);
    float s1 = fmaf(wx, b1.y - b1.x, b1.x);
    vy = fmaf(wy, s1 - s0, s0);
}

// One thread per pixel: integrate pred & true trajectories together,
// accumulate squared trajectory difference, block-reduce to one partial.
__global__ __launch_bounds__(256) void ivp_traj_kernel(
        const float* __restrict__ vf_pred,
        const float* __restrict__ vf_true,
        float* __restrict__ partials) {
    int idx = blockIdx.x * 256 + threadIdx.x;          // 0 .. B*H*W-1
    int b = idx >> 18;                                  // / (512*512)
    int rem = idx & (HW - 1);
    float px = (float)(rem & 511);                      // x along W
    float py = (float)(rem >> 9);                       // y along H
    const float* bp = vf_pred + (size_t)b * (2 * HW);
    const float* bt = vf_true + (size_t)b * (2 * HW);
    float ppx = px, ppy = py;   // pred trajectory (unclipped position)
    float ptx = px, pty = py;   // true trajectory
    float acc = 0.0f;
#pragma unroll
    for (int s = 0; s < NSTEPS; ++s) {
        float vx, vy, ux, uy;
        bilerp2(bp, ppx, ppy, vx, vy);   // 8 independent b64 gathers in flight
        bilerp2(bt, ptx, pty, ux, uy);
        ppx = fmaf(DXC, vx, ppx);
        ppy = fmaf(DXC, vy, ppy);
        ptx = fmaf(DXC, ux, ptx);
        pty = fmaf(DXC, uy, pty);
        float ddx = ppx - ptx, ddy = ppy - pty;
        acc = fmaf(ddx, ddx, acc);
        acc = fmaf(ddy, ddy, acc);
    }
    // wave32 butterfly reduction, then one cross-wave LDS pass
#pragma unroll
    for (int m = 16; m >= 1; m >>= 1) acc += __shfl_xor(acc, m, 32);
    __shared__ float wsum[8];
    int wid = threadIdx.x >> 5;
    int lane = threadIdx.x & 31;
    if (lane == 0) wsum[wid] = acc;
    __syncthreads();
    if (threadIdx.x == 0) {
        float s = 0.0f;
#pragma unroll
        for (int w = 0; w < 8; ++w) s += wsum[w];
        partials[blockIdx.x] = s;
    }
}

// Deterministic final reduction of NPARTIALS partials.
// 1024 threads absorb 8 values each via two b128 loads (latency overlapped
// across 32 waves), LDS tree down to 32, then wave 0 finishes with
// V_WMMA_F32_16X16X4_F32:
//  WMMA#1: A holds one distinct per-lane partial per A-slot (other slot 0),
//          B = ones  ->  D rows are partial row-sums, constant across N.
//  WMMA#2..5: A = ones, B = pairs of D VGPRs (each half-wave of a B VGPR is
//          a constant row) -> every element of D2 becomes the grand total.
// Robust to A/B row-ordering details; relies only on documented striping.
__global__ __launch_bounds__(1024) void ivp_reduce_kernel(
        const float* __restrict__ partials, float* __restrict__ out) {
    int tid = threadIdx.x;
    v4f x0 = *(const v4f*)(partials + 4 * tid);                    // 0..4095
    v4f x1 = *(const v4f*)(partials + 4 * tid + NPARTIALS / 2);    // 4096..8191
    float acc = ((x0.x + x0.y) + (x0.z + x0.w)) +
                ((x1.x + x1.y) + (x1.z + x1.w));
    __shared__ float red[1024];
    red[tid] = acc;
    __syncthreads();
#pragma unroll
    for (int off = 512; off >= 32; off >>= 1) {
        if (tid < off) red[tid] += red[tid + off];
        __syncthreads();
    }
    if (tid < 32) {  // exactly wave 0, EXEC all ones
        v2f a;    a.x = red[tid]; a.y = 0.0f;
        v2f ones; ones.x = 1.0f;  ones.y = 1.0f;
        v8f c = {};
        v8f d1 = __builtin_amdgcn_wmma_f32_16x16x4_f32(
            false, a, false, ones, (short)0, c, false, false);

        v8f d2 = {};
        v2f bfrag;
        bfrag.x = d1[0]; bfrag.y = d1[1];
        d2 = __builtin_amdgcn_wmma_f32_16x16x4_f32(
            false, ones, false, bfrag, (short)0, d2, false, false);
        bfrag.x = d1[2]; bfrag.y = d1[3];
        d2 = __builtin_amdgcn_wmma_f32_16x16x4_f32(
            false, ones, false, bfrag, (short)0, d2, false, false);
        bfrag.x = d1[4]; bfrag.y = d1[5];
        d2 = __builtin_amdgcn_wmma_f32_16x16x4_f32(
            false, ones, false, bfrag, (short)0, d2, false, false);
        bfrag.x = d1[6]; bfrag.y = d1[7];
        d2 = __builtin_amdgcn_wmma_f32_16x16x4_f32(
            false, ones, false, bfrag, (short)0, d2, false, false);

        if (tid == 0) out[0] = d2[0] * INV_COUNT;
    }
}

extern "C" void kernel_launch(void* const* d_in, const int* in_sizes, int n_in,
                              void* d_out, int out_size, void* d_ws, size_t ws_size,
                              hipStream_t stream) {
    (void)in_sizes; (void)n_in; (void)out_size; (void)ws_size;
    const float* vf_pred = (const float*)d_in[0];
    const float* vf_true = (const float*)d_in[1];
    float* partials = (float*)d_ws;            // 8192 floats = 32 KB scratch
    float* out = (float*)d_out;

    const int npix = NB * HW;                  // 2,097,152
    const int nblocks = npix / 256;            // 8192 == NPARTIALS

    ivp_traj_kernel<<<nblocks, 256, 0, stream>>>(vf_pred, vf_true, partials);
    ivp_reduce_kernel<<<1, 1024, 0, stream>>>(partials, out);
}